// RenderLayer_67310727463523
// MI455X (gfx1250) — compile-verified
//
#include <hip/hip_runtime.h>
#include <math.h>

#define BATCH 2
#define VN    5000
#define FC    10000
#define HH    256
#define WW    256
#define TW    1024
#define HWP   (HH*WW)

typedef float v2f __attribute__((ext_vector_type(2)));
typedef float v8f __attribute__((ext_vector_type(8)));

// ---------------------------------------------------------------------------
// Kernel 1: vertex transform  v_pix = project(K, Rt, verts)
// ---------------------------------------------------------------------------
__global__ void k_transform(const float* __restrict__ verts,
                            const float* __restrict__ Km_,
                            const float* __restrict__ Rt_,
                            float* __restrict__ vpix) {
  int i = blockIdx.x * blockDim.x + threadIdx.x;
  if (i >= BATCH * VN) return;
  int b = i / VN;
  float x = verts[i*3+0], y = verts[i*3+1], z = verts[i*3+2];
  const float* R = Rt_ + b*12;
  float cxv = R[0]*x + R[1]*y + R[2]*z  + R[3];
  float cyv = R[4]*x + R[5]*y + R[6]*z  + R[7];
  float czv = R[8]*x + R[9]*y + R[10]*z + R[11];
  const float* K = Km_ + b*9;
  float px = K[0]*cxv + K[1]*cyv + K[2]*czv;
  float py = K[3]*cxv + K[4]*cyv + K[5]*czv;
  vpix[i*3+0] = px / czv;
  vpix[i*3+1] = py / czv;
  vpix[i*3+2] = czv;
}

// ---------------------------------------------------------------------------
// Kernel 2: per-face affine plane coefficients (edge fns pre-scaled by sign,
// plus the inverse-depth plane). Invalid faces get C=-1e30 so inside == false.
// Layout per (b,face): 4 planes x float4 {Ax, Ay, C, 0} = 16 floats.
// ---------------------------------------------------------------------------
__global__ void k_face_planes(const float* __restrict__ vpix,
                              const int* __restrict__ vi,
                              float* __restrict__ planes) {
  int i = blockIdx.x * blockDim.x + threadIdx.x;
  if (i >= BATCH * FC) return;
  int b = i / FC, f = i - b*FC;
  const float* vb = vpix + (size_t)b*VN*3;
  int i0 = vi[f*3+0], i1 = vi[f*3+1], i2 = vi[f*3+2];
  float ax=vb[i0*3], ay=vb[i0*3+1], az=vb[i0*3+2];
  float bx=vb[i1*3], by=vb[i1*3+1], bz=vb[i1*3+2];
  float cx=vb[i2*3], cy=vb[i2*3+1], cz=vb[i2*3+2];
  float area = (bx-ax)*(cy-ay) - (by-ay)*(cx-ax);
  float s = (area > 0.f) ? 1.f : ((area < 0.f) ? -1.f : 0.f);
  bool  nz = fabsf(area) > 1e-8f;
  bool  valid = nz && (az > 0.f) && (bz > 0.f) && (cz > 0.f);
  float ar = nz ? area : 1.0f;
  // w0 = cross2(b, c, p), w1 = cross2(c, a, p), w2 = cross2(a, b, p)
  float e0x = by-cy, e0y = cx-bx, e0c = bx*cy - by*cx;
  float e1x = cy-ay, e1y = ax-cx, e1c = cx*ay - cy*ax;
  float e2x = ay-by, e2y = bx-ax, e2c = ax*by - ay*bx;
  float* P = planes + (size_t)i*16;
  if (valid) {
    float k0 = 1.0f/(ar*az), k1 = 1.0f/(ar*bz), k2 = 1.0f/(ar*cz);
    P[0]=s*e0x;  P[1]=s*e0y;  P[2]=s*e0c;  P[3]=0.f;
    P[4]=s*e1x;  P[5]=s*e1y;  P[6]=s*e1c;  P[7]=0.f;
    P[8]=s*e2x;  P[9]=s*e2y;  P[10]=s*e2c; P[11]=0.f;
    P[12]=e0x*k0+e1x*k1+e2x*k2;
    P[13]=e0y*k0+e1y*k1+e2y*k2;
    P[14]=e0c*k0+e1c*k1+e2c*k2;
    P[15]=0.f;
  } else {
    #pragma unroll
    for (int j = 0; j < 16; ++j) P[j] = 0.f;
    P[2] = -1e30f; P[6] = -1e30f; P[10] = -1e30f;   // never inside
  }
}

// ---------------------------------------------------------------------------
// Kernel 3: WMMA rasterizer. Each wave owns 16 pixels (N dim); each
// V_WMMA_F32_16X16X4_F32 evaluates one affine plane for 16 faces x 16 pixels.
// 4 WMMAs per 16-face group (w0', w1', w2', invz). Depth test = max invz
// (bestz >= 0 invariant makes the invz>0 test implicit); ties -> smallest
// face id (matches the reference scan semantics). Inside test uses the
// sign-bit OR trick: all three edge values non-negative <=> OR of their IEEE
// bit patterns is non-negative as an int.
// ---------------------------------------------------------------------------
__global__ __launch_bounds__(256) void k_raster(const float* __restrict__ planes,
                                                int* __restrict__ idx_ws,
                                                float* __restrict__ idx_out) {
  const int b    = blockIdx.z;
  const int lane = threadIdx.x & 31;
  const int wave = threadIdx.x >> 5;
  const int xn   = lane & 15;
  const int half = lane >> 4;
  const int y    = blockIdx.y * 8 + wave;
  const int x    = blockIdx.x * 16 + xn;

  // B operand 4x16: K0=px K1=py (lanes 0-15), K2=1 K3=0 (lanes 16-31)
  v2f bm;
  if (half == 0) { bm.x = (float)x + 0.5f; bm.y = (float)y + 0.5f; }
  else           { bm.x = 1.0f;            bm.y = 0.0f;            }

  float bestz = 0.0f;   // running max of invz; invariant bestz >= 0
  int   besti = -1;
  // A operand 16x4: lanes 0-15 -> (Ax,Ay) of face base+m; lanes 16-31 -> (C,0)
  const v2f* fp = (const v2f*)(planes + (size_t)b * FC * 16) + (size_t)xn * 8 + half;
  int fidbase = half * 8;

  #pragma unroll 2
  for (int g = 0; g < FC/16; ++g) {
    v2f A0 = fp[0];
    v2f A1 = fp[2];
    v2f A2 = fp[4];
    v2f A3 = fp[6];
    v8f cacc = {};
    v8f d0 = __builtin_amdgcn_wmma_f32_16x16x4_f32(false, A0, false, bm, (short)0, cacc, false, false);
    v8f d1 = __builtin_amdgcn_wmma_f32_16x16x4_f32(false, A1, false, bm, (short)0, cacc, false, false);
    v8f d2 = __builtin_amdgcn_wmma_f32_16x16x4_f32(false, A2, false, bm, (short)0, cacc, false, false);
    v8f dz = __builtin_amdgcn_wmma_f32_16x16x4_f32(false, A3, false, bm, (short)0, cacc, false, false);
    #pragma unroll
    for (int r = 0; r < 8; ++r) {
      float iz = dz[r];
      int sgn = __float_as_int(d0[r]) | __float_as_int(d1[r]) | __float_as_int(d2[r]);
      bool take = (sgn >= 0) && (iz > bestz);
      bestz = take ? iz : bestz;
      besti = take ? (fidbase + r) : besti;
    }
    fp      += 16 * 8;   // next 16-face group (16 faces x 8 v2f)
    fidbase += 16;
  }
  // merge the two half-wave views of the same pixel (wave32 shuffle)
  float oz = __shfl_xor(bestz, 16, 32);
  int   oi = __shfl_xor(besti, 16, 32);
  if ((oz > bestz) || (oz == bestz && oi >= 0 && oi < besti)) { bestz = oz; besti = oi; }
  if (half == 0) {
    int o = (b*HH + y)*WW + x;
    idx_ws[o]  = besti;
    idx_out[o] = (float)besti;
  }
}

// ---------------------------------------------------------------------------
// Kernel 4: shade — barycentrics, depth, uv interpolation, bilinear texture
// ---------------------------------------------------------------------------
__global__ void k_render(const float* __restrict__ vpix,
                         const int* __restrict__ vi,
                         const int* __restrict__ vti,
                         const float* __restrict__ vt,
                         const float* __restrict__ tex,
                         const int* __restrict__ idx_ws,
                         float* __restrict__ o_img,  float* __restrict__ o_depth,
                         float* __restrict__ o_vt,   float* __restrict__ o_bary,
                         float* __restrict__ o_mask) {
  int i = blockIdx.x * blockDim.x + threadIdx.x;
  if (i >= BATCH * HWP) return;
  int b = i / HWP;
  int p = i - b * HWP;
  int y = p / WW, x = p - y * WW;
  int idx = idx_ws[i];
  float m = (idx != -1) ? 1.0f : 0.0f;
  int ic = idx < 0 ? 0 : idx;
  const float* vb = vpix + (size_t)b*VN*3;
  int i0 = vi[ic*3+0], i1 = vi[ic*3+1], i2 = vi[ic*3+2];
  float ax=vb[i0*3], ay=vb[i0*3+1], az=vb[i0*3+2];
  float bx=vb[i1*3], by=vb[i1*3+1], bz=vb[i1*3+2];
  float cx=vb[i2*3], cy=vb[i2*3+1], cz=vb[i2*3+2];
  float ptx = (float)x + 0.5f, pty = (float)y + 0.5f;
  float w0 = (cx-bx)*(pty-by) - (cy-by)*(ptx-bx);
  float w1 = (ax-cx)*(pty-cy) - (ay-cy)*(ptx-cx);
  float w2 = (bx-ax)*(pty-ay) - (by-ay)*(ptx-ax);
  float area = w0 + w1 + w2;
  float arr = (fabsf(area) > 1e-8f) ? area : 1.0f;
  float f0 = w0 / arr / fmaxf(az, 1e-8f);
  float f1 = w1 / arr / fmaxf(bz, 1e-8f);
  float f2 = w2 / arr / fmaxf(cz, 1e-8f);
  float invz = f0 + f1 + f2;
  float invz_s = (fabsf(invz) > 1e-12f) ? invz : 1.0f;
  float depth = 1.0f / invz_s;
  float b0 = f0 / invz_s * m;
  float b1 = f1 / invz_s * m;
  float b2 = f2 / invz_s * m;
  o_depth[i] = depth * m;
  o_bary[((size_t)b*3+0)*HWP + p] = b0;
  o_bary[((size_t)b*3+1)*HWP + p] = b1;
  o_bary[((size_t)b*3+2)*HWP + p] = b2;
  int t0 = vti[ic*3+0], t1 = vti[ic*3+1], t2 = vti[ic*3+2];
  float u0 = vt[t0*2+0]*2.f-1.f, v0 = vt[t0*2+1]*2.f-1.f;
  float u1 = vt[t1*2+0]*2.f-1.f, v1 = vt[t1*2+1]*2.f-1.f;
  float u2 = vt[t2*2+0]*2.f-1.f, v2 = vt[t2*2+1]*2.f-1.f;
  float gx = u0*b0 + u1*b1 + u2*b2;
  float gy = v0*b0 + v1*b1 + v2*b2;
  o_vt[((size_t)b*2+0)*HWP + p] = gx;
  o_vt[((size_t)b*2+1)*HWP + p] = gy;
  float sx = (gx + 1.0f) * (TW * 0.5f) - 0.5f;
  float sy = (gy + 1.0f) * (TW * 0.5f) - 0.5f;
  float x0f = floorf(sx), y0f = floorf(sy);
  float wx = sx - x0f, wy = sy - y0f;
  int x0 = (int)x0f, y0 = (int)y0f;
  const float* tb = tex + (size_t)b*3*TW*TW;
  float acc0 = 0.f, acc1 = 0.f, acc2 = 0.f;
  #pragma unroll
  for (int t = 0; t < 4; ++t) {
    int ix = x0 + (t & 1);
    int iy = y0 + (t >> 1);
    float wt = ((t & 1) ? wx : 1.f - wx) * ((t >> 1) ? wy : 1.f - wy);
    bool v = (ix >= 0) && (ix < TW) && (iy >= 0) && (iy < TW);
    int ixc = ix < 0 ? 0 : (ix > TW-1 ? TW-1 : ix);
    int iyc = iy < 0 ? 0 : (iy > TW-1 ? TW-1 : iy);
    float vf = v ? wt : 0.0f;
    size_t off = (size_t)iyc*TW + ixc;
    acc0 += tb[(size_t)0*TW*TW + off] * vf;
    acc1 += tb[(size_t)1*TW*TW + off] * vf;
    acc2 += tb[(size_t)2*TW*TW + off] * vf;
  }
  o_img[((size_t)b*3+0)*HWP + p] = acc0 * m;
  o_img[((size_t)b*3+1)*HWP + p] = acc1 * m;
  o_img[((size_t)b*3+2)*HWP + p] = acc2 * m;
  o_mask[i] = m;
}

// ---------------------------------------------------------------------------
extern "C" void kernel_launch(void* const* d_in, const int* in_sizes, int n_in,
                              void* d_out, int out_size, void* d_ws, size_t ws_size,
                              hipStream_t stream) {
  const float* verts = (const float*)d_in[0];
  const float* tex   = (const float*)d_in[1];
  const float* K     = (const float*)d_in[2];
  const float* Rt    = (const float*)d_in[3];
  const float* vt    = (const float*)d_in[4];
  const int*   vi    = (const int*)d_in[5];
  const int*   vti   = (const int*)d_in[6];

  float* out     = (float*)d_out;
  float* o_img   = out;                                   // B*3*H*W
  float* o_depth = o_img  + (size_t)BATCH*3*HWP;          // B*H*W
  float* o_vpix  = o_depth + (size_t)BATCH*HWP;           // B*V*3
  float* o_vt    = o_vpix + (size_t)BATCH*VN*3;           // B*2*H*W
  float* o_idx   = o_vt   + (size_t)BATCH*2*HWP;          // B*H*W (float cast)
  float* o_bary  = o_idx  + (size_t)BATCH*HWP;            // B*3*H*W
  float* o_mask  = o_bary + (size_t)BATCH*3*HWP;          // B*H*W

  float* planes = (float*)d_ws;                                     // B*FC*16 floats
  int*   idx_ws = (int*)((char*)d_ws + (size_t)BATCH*FC*16*sizeof(float));

  k_transform<<<dim3((BATCH*VN + 255)/256), dim3(256), 0, stream>>>(verts, K, Rt, o_vpix);
  k_face_planes<<<dim3((BATCH*FC + 255)/256), dim3(256), 0, stream>>>(o_vpix, vi, planes);
  k_raster<<<dim3(WW/16, HH/8, BATCH), dim3(256), 0, stream>>>(planes, idx_ws, o_idx);
  k_render<<<dim3((BATCH*HWP + 255)/256), dim3(256), 0, stream>>>(
      o_vpix, vi, vti, vt, tex, idx_ws, o_img, o_depth, o_vt, o_bary, o_mask);
}